// Linear_52175262711977
// MI455X (gfx1250) — compile-verified
//
#include <hip/hip_runtime.h>
#include <hip/hip_bf16.h>

// Equivariant block-diagonal linear (e3nn "Linear" on 128x0e+128x1o+128x2e):
//   per irrep l (d = 2l+1):  y[n,w,i] = sum_u x[n,u,i] * W_l[u,w]
// == fp32 GEMM  [N*d, 128] x [128, 128]  via v_wmma_f32_16x16x4_f32, with the
// per-block x tile staged into LDS by the Tensor Data Mover (tensor_load_to_lds).

typedef __attribute__((ext_vector_type(2))) float        v2f;
typedef __attribute__((ext_vector_type(8))) float        v8f;
typedef __attribute__((ext_vector_type(4))) unsigned int v4u;
typedef __attribute__((ext_vector_type(8))) int          v8i;
typedef __attribute__((ext_vector_type(4))) int          v4i;

#define TOTAL_DIM 1152
#define NNODES    50000

// One block: 8 waves, 128 rows (M) of one irrep's GEMM, full N=128 output cols.
// Wave w owns output columns [16w, 16w+16) and keeps that column strip of W
// resident in 64 VGPRs (32 K-chunks x v2f) for the whole block.
template <int D, int OFF>
__global__ __launch_bounds__(256)
void irrep_linear_wmma(const float* __restrict__ x,
                       const float* __restrict__ W,
                       float* __restrict__ out,
                       int M /* = NNODES * D, rows of this irrep's GEMM */) {
    constexpr int CHUNK = 128 * D;                 // floats per node for this irrep
    constexpr int MAXNODES = (127 / D) + 2;        // nodes a 128-row block can span
    __shared__ float xs[MAXNODES * CHUNK];         // sole LDS object -> offset 0

    const int tid    = threadIdx.x;
    const int lane   = tid & 31;
    const int waveId = tid >> 5;

    const int R0 = blockIdx.x * 128;               // first M-row of this block
    const int n0 = R0 / D;                         // first node touched
    const int rEnd = min(R0 + 127, M - 1);
    const int cnt  = rEnd / D - n0 + 1;            // nodes spanned

    // ---- Stage this block's x rows into LDS.  The region is a 2D tile:
    // cnt rows of CHUNK contiguous floats at row stride TOTAL_DIM floats —
    // exactly one Tensor Data Mover descriptor.  Issue from wave 0 (TDM
    // ignores EXEC; TENSORcnt is per-wave), everyone else meets the barrier.
#if __has_builtin(__builtin_amdgcn_tensor_load_to_lds) && defined(__gfx1250__)
    if (waveId == 0) {
        const unsigned long long g =
            (unsigned long long)(const void*)x +
            ((unsigned long long)n0 * TOTAL_DIM + OFF) * 4ull;
        v4u g0;
        g0[0] = 1u;                                          // count=1 (valid, user mode)
        g0[1] = 0u;                                          // lds_addr = 0
        g0[2] = (unsigned)(g & 0xffffffffu);                 // global_addr[31:0]
        g0[3] = ((unsigned)(g >> 32) & 0x01ffffffu) | (2u << 30); // ga[56:32] | type=2
        v8i g1;
        g1[0] = (int)(2u << 16);                             // data_size = 4B
        g1[1] = (int)((unsigned)CHUNK << 16);                // tensor_dim0[15:0]
        g1[2] = (int)((unsigned)cnt << 16);                  // tensor_dim1[15:0]
        g1[3] = (int)((unsigned)CHUNK << 16);                // tile_dim0
        g1[4] = (int)(unsigned)cnt;                          // tile_dim1 (tile_dim2=0)
        g1[5] = (int)TOTAL_DIM;                              // tensor_dim0_stride[31:0]
        g1[6] = 0;                                           // stride hi / dim1_stride lo
        g1[7] = 0;
        const v4i gz4 = {0, 0, 0, 0};                        // 2D tile: groups 2/3 unused
        const v8i gz8 = {0, 0, 0, 0, 0, 0, 0, 0};
        __builtin_amdgcn_tensor_load_to_lds(g0, g1, gz4, gz4, gz8, 0);
        __builtin_amdgcn_s_wait_tensorcnt(0);
    }
#else
    {
        const int F = cnt * CHUNK;                 // floats to stage (mult of 4)
        for (int f = tid * 4; f < F; f += 256 * 4) {
            const int node  = f / CHUNK;
            const int local = f - node * CHUNK;
            const float4 v = *reinterpret_cast<const float4*>(
                x + (size_t)(n0 + node) * TOTAL_DIM + OFF + local);
            *reinterpret_cast<float4*>(&xs[f]) = v;
        }
    }
#endif

    // ---- Per-wave B fragments: column strip [16*waveId, 16*waveId+16) of W.
    // 32-bit B 4x16 layout: VGPR0 = rows {K, K+2} (lane halves), VGPR1 = {K+1, K+3}.
    const int nbase = waveId * 16;
    const int col   = nbase + (lane & 15);
    const int ksel  = (lane >> 4) << 1;            // 0 for lanes 0-15, 2 for 16-31
    v2f bfrag[32];
#pragma unroll
    for (int kc = 0; kc < 32; ++kc) {
        const int k = kc * 4 + ksel;
        bfrag[kc].x = W[k * 128 + col];
        bfrag[kc].y = W[(k + 1) * 128 + col];
    }

    __syncthreads();                               // LDS tile ready for all waves

    // ---- Compute: 8 M-tiles of 16 rows, accumulate over K=128 in chunks of 4.
    const int mr = lane & 15;                      // A-matrix M row within tile
    for (int mt = 0; mt < 8; ++mt) {
        const int Rm = R0 + mt * 16;
        if (Rm >= M) break;                        // uniform guard (last block)

        // A-frag source row in LDS: row r -> node r/D, component r%D.
        const int r  = Rm + mr;
        const int n  = r / D;
        const int i  = r - n * D;
        const float* arow = &xs[(n - n0) * CHUNK + i];

        v8f c = {};                                // C = 0
#pragma unroll
        for (int kc = 0; kc < 32; ++kc) {
            const int k = kc * 4 + ksel;
            v2f a;                                 // 32-bit A 16x4 layout
            a.x = arow[k * D];
            a.y = arow[(k + 1) * D];
            c = __builtin_amdgcn_wmma_f32_16x16x4_f32(
                    /*neg_a=*/false, a, /*neg_b=*/false, bfrag[kc],
                    /*c_mod=*/(short)0, c, /*reuse_a=*/false, /*reuse_b=*/false);
        }

        // 32-bit C/D 16x16 layout: VGPR j holds M=j (lanes 0-15) / M=j+8 (16-31).
        const int rhalf = (lane >> 4) << 3;        // 0 or 8
#pragma unroll
        for (int j = 0; j < 8; ++j) {
            const int rr = Rm + j + rhalf;
            const int nn = rr / D;
            const int ii = rr - nn * D;
            out[(size_t)nn * TOTAL_DIM + OFF + col * D + ii] = c[j];
        }
    }
}

extern "C" void kernel_launch(void* const* d_in, const int* in_sizes, int n_in,
                              void* d_out, int out_size, void* d_ws, size_t ws_size,
                              hipStream_t stream) {
    const float* x  = (const float*)d_in[0];
    const float* w0 = (const float*)d_in[1];
    const float* w1 = (const float*)d_in[2];
    const float* w2 = (const float*)d_in[3];
    float* out = (float*)d_out;

    constexpr int M0 = NNODES * 1;   // 50000
    constexpr int M1 = NNODES * 3;   // 150000
    constexpr int M2 = NNODES * 5;   // 250000

    dim3 blk(256);
    irrep_linear_wmma<1, 0>  <<<dim3((M0 + 127) / 128), blk, 0, stream>>>(x, w0, out, M0);
    irrep_linear_wmma<3, 128><<<dim3((M1 + 127) / 128), blk, 0, stream>>>(x, w1, out, M1);
    irrep_linear_wmma<5, 512><<<dim3((M2 + 127) / 128), blk, 0, stream>>>(x, w2, out, M2);
}